// SoftWindow_36876589204077
// MI455X (gfx1250) — compile-verified
//
#include <hip/hip_runtime.h>
#include <cstdint>

typedef __attribute__((ext_vector_type(2))) float v2f;
typedef __attribute__((ext_vector_type(8))) float v8f;

#define B_ 256
#define U_ 4096
#define V_ 80
#define H_ 1024
#define K_ 10
#define SPLITS 16          // u-splits per (b, vtile); 256 u each
#define VT 5               // 5 v-tiles of 16 cover V=80
#define PT_STRIDE (U_ + 1) // 4097

// d_out layout (floats), concatenated in return order (w, new_kappa, phi_termination)
#define OUT_W_OFF  0
#define OUT_NK_OFF (B_ * V_)            // 20480
#define OUT_PT_OFF (B_ * V_ + B_ * K_)  // 23040

// d_ws layout (floats)
#define WS_PART_OFF  0                          // B*SPLITS*V partial sums (1.31 MB)
#define WS_ALPHA_OFF (B_ * SPLITS * V_)
#define WS_BETA_OFF  (WS_ALPHA_OFF + B_ * K_)
#define WS_NK_OFF    (WS_BETA_OFF + B_ * K_)

// ---------------------------------------------------------------------------
// Kernel A: lin = x @ W^T + b ; alpha/beta/new_kappa. One wave32 per (b, j).
// ---------------------------------------------------------------------------
__global__ void SoftWindow_lin_kernel(const float* __restrict__ x,
                                      const float* __restrict__ W,
                                      const float* __restrict__ bias,
                                      const float* __restrict__ kappa,
                                      float* __restrict__ out,
                                      float* __restrict__ ws) {
    int wave = (blockIdx.x * blockDim.x + threadIdx.x) >> 5;
    int lane = threadIdx.x & 31;
    if (wave >= B_ * 3 * K_) return;
    int b = wave / (3 * K_);
    int j = wave % (3 * K_);

    const float* xr = x + (size_t)b * H_;
    const float* wr = W + (size_t)j * H_;
    float s = 0.f;
    #pragma unroll 4
    for (int h = lane; h < H_; h += 32)
        s = __builtin_fmaf(xr[h], wr[h], s);
    #pragma unroll
    for (int off = 16; off > 0; off >>= 1)
        s += __shfl_down(s, off, 32);

    if (lane == 0) {
        s += bias[j];
        if (j < K_) {
            ws[WS_ALPHA_OFF + b * K_ + j] = expf(s);
        } else if (j < 2 * K_) {
            ws[WS_BETA_OFF + b * K_ + (j - K_)] = expf(s);
        } else {
            int k = j - 2 * K_;
            float nk = kappa[b * K_ + k] + expf(s - 3.9f);
            ws[WS_NK_OFF + b * K_ + k] = nk;
            out[OUT_NK_OFF + b * K_ + k] = nk;
        }
    }
}

// ---------------------------------------------------------------------------
// Kernel B: phi_termination[b,u] = sum_k alpha*exp(-beta*(nk-u)^2), u in [0,U]
// ---------------------------------------------------------------------------
__global__ void SoftWindow_phi_kernel(const float* __restrict__ ws,
                                      float* __restrict__ out) {
    int i = blockIdx.x * blockDim.x + threadIdx.x;
    if (i >= B_ * PT_STRIDE) return;
    int b = i / PT_STRIDE;
    int u = i - b * PT_STRIDE;

    const float* al = ws + WS_ALPHA_OFF + b * K_;
    const float* be = ws + WS_BETA_OFF + b * K_;
    const float* nk = ws + WS_NK_OFF + b * K_;
    float uu = (float)u;
    float s = 0.f;
    #pragma unroll
    for (int k = 0; k < K_; ++k) {
        float d = nk[k] - uu;
        s += al[k] * expf(-be[k] * d * d);
    }
    out[OUT_PT_OFF + i] = s;
}

// ---------------------------------------------------------------------------
// Kernel C: w[b,v] = sum_u phi_masked[b,u] * c[b,u,v] via V_WMMA_F32_16X16X4_F32.
// One wave32 per (b, vtile, usplit): A row 0 = 4 phi values, B = 4x16 c tile.
// All loop bounds are forced into SGPRs via readfirstlane (they are wave-
// uniform by construction), so loop control is scalar and EXEC stays all-ones
// around the WMMAs. phi[b,u]==0 for u >= len, so the u-loop is clamped to len
// (skips ~half the 335 MB of c traffic on average). Head-lane masking of the
// A matrix is a bitwise AND -> branchless.
// ---------------------------------------------------------------------------
__global__ void SoftWindow_window_wmma_kernel(const float* __restrict__ c,
                                              const int* __restrict__ lens,
                                              const float* __restrict__ out_pt,
                                              float* __restrict__ partials) {
    int gwave = (blockIdx.x * blockDim.x + threadIdx.x) >> 5;
    gwave = __builtin_amdgcn_readfirstlane(gwave);      // wave-uniform -> SGPR
    int lane  = threadIdx.x & 31;
    int s  = gwave % SPLITS;
    int vt = (gwave / SPLITS) % VT;
    int b  = gwave / (SPLITS * VT);
    if (b >= B_) return;

    const int ulen = U_ / SPLITS;             // 256
    const int u0   = s * ulen;
    const int uend = u0 + ulen;
    const int col  = vt * 16 + (lane & 15);   // output v index for this lane
    const int koff = (lane >> 4) << 1;        // lanes 0-15 -> K0/K1, lanes 16-31 -> K2/K3
    const uint32_t headmask = ((lane & 15) == 0) ? 0xFFFFFFFFu : 0u; // only M=0 row of A
    const int len = __builtin_amdgcn_readfirstlane(lens[b]);         // uniform -> SGPR

    const float* crow = c + (size_t)b * U_ * V_ + col;   // c[b, u, col] = crow[u*V_]
    const float* phi  = out_pt + (size_t)b * PT_STRIDE;

    // Fast region: all 4 rows of each step are < len -> no len-masking needed.
    int fast_end = len & ~3;
    if (fast_end < u0)   fast_end = u0;
    if (fast_end > uend) fast_end = uend;

    v8f acc = {};
    #pragma unroll 4
    for (int u = u0; u < fast_end; u += 4) {
        int r0 = u + koff;
        int r1 = r0 + 1;

        v2f a;
        a.x = __uint_as_float(__float_as_uint(phi[r0]) & headmask);
        a.y = __uint_as_float(__float_as_uint(phi[r1]) & headmask);

        v2f bm;
        bm.x = crow[(size_t)r0 * V_];
        bm.y = crow[(size_t)r1 * V_];

        acc = __builtin_amdgcn_wmma_f32_16x16x4_f32(
            false, a, false, bm, (short)0, acc, false, false);
    }

    // Boundary step (at most one): rows fast_end..fast_end+3 straddle len.
    if (fast_end < uend && fast_end < len) {
        int r0 = fast_end + koff;
        int r1 = r0 + 1;
        uint32_t m0 = (r0 < len) ? headmask : 0u;
        uint32_t m1 = (r1 < len) ? headmask : 0u;

        v2f a;
        a.x = __uint_as_float(__float_as_uint(phi[r0]) & m0);
        a.y = __uint_as_float(__float_as_uint(phi[r1]) & m1);

        v2f bm;
        bm.x = crow[(size_t)r0 * V_];
        bm.y = crow[(size_t)r1 * V_];

        acc = __builtin_amdgcn_wmma_f32_16x16x4_f32(
            false, a, false, bm, (short)0, acc, false, false);
    }

    // D row M=0 lives in VGPR0 (acc[0]) of lanes 0..15. Always write (zeros too).
    if (lane < 16)
        partials[((size_t)b * SPLITS + s) * V_ + vt * 16 + lane] = acc[0];
}

// ---------------------------------------------------------------------------
// Kernel D: reduce split-K partials -> w
// ---------------------------------------------------------------------------
__global__ void SoftWindow_reduce_kernel(const float* __restrict__ partials,
                                         float* __restrict__ out_w) {
    int i = blockIdx.x * blockDim.x + threadIdx.x;
    if (i >= B_ * V_) return;
    int b = i / V_;
    int v = i - b * V_;
    float s = 0.f;
    #pragma unroll
    for (int sp = 0; sp < SPLITS; ++sp)
        s += partials[((size_t)b * SPLITS + sp) * V_ + v];
    out_w[i] = s;
}

extern "C" void kernel_launch(void* const* d_in, const int* in_sizes, int n_in,
                              void* d_out, int out_size, void* d_ws, size_t ws_size,
                              hipStream_t stream) {
    const float* x     = (const float*)d_in[0];
    const float* c     = (const float*)d_in[1];
    const float* kappa = (const float*)d_in[2];
    const float* W     = (const float*)d_in[3];
    const float* bias  = (const float*)d_in[4];
    const int*   lens  = (const int*)d_in[5];
    float* out = (float*)d_out;
    float* ws  = (float*)d_ws;

    // A: 256*30 waves, 8 waves per 256-thread block -> 960 blocks
    SoftWindow_lin_kernel<<<960, 256, 0, stream>>>(x, W, bias, kappa, out, ws);

    // B: 256*4097 threads
    int nPhi = B_ * PT_STRIDE;
    SoftWindow_phi_kernel<<<(nPhi + 255) / 256, 256, 0, stream>>>(ws, out);

    // C: 256*5*16 = 20480 waves -> 2560 blocks of 8 waves
    int nWaves = B_ * VT * SPLITS;
    SoftWindow_window_wmma_kernel<<<nWaves / 8, 256, 0, stream>>>(
        c, lens, out + OUT_PT_OFF, ws + WS_PART_OFF);

    // D: final reduction into w
    SoftWindow_reduce_kernel<<<(B_ * V_ + 255) / 256, 256, 0, stream>>>(
        ws + WS_PART_OFF, out + OUT_W_OFF);
}